// YOLOLoss_87084756894179
// MI455X (gfx1250) — compile-verified
//
#include <hip/hip_runtime.h>
#include <hip/hip_bf16.h>

typedef float v2f __attribute__((ext_vector_type(2)));
typedef float v8f __attribute__((ext_vector_type(8)));

#define L_COORD 5.0f
#define L_NOOBJ 1.0f
#define L_CLS   1.0f

// Problem geometry (fixed by setup_inputs)
constexpr int Bq = 32, Aq = 3, Hq = 128, Wq = 128, NCq = 10, Nq = 50;
constexpr int Cq = 5 + NCq;                 // 15 channels
constexpr int Mcells = Bq * Aq * Hq * Wq;   // 1,572,864
constexpr int NT = Bq * Nq;                 // 1600 targets

constexpr int TPB = 256;
constexpr int BLOCKS = 1024;                // 262144 threads -> 6 cells/thread
static_assert(Mcells % (TPB * BLOCKS) == 0, "even split");
constexpr int CHUNKS_P = BLOCKS / 64;       // 16 WMMA chunks for block partials
constexpr int CHUNKS_T = NT / 64;           // 25 WMMA chunks per contribution array
static_assert(BLOCKS % 64 == 0 && NT % 64 == 0, "64-float WMMA chunks");

// Workspace layout (floats/ints)
//   [0, BLOCKS)                 : per-block softplus partials
//   [BLOCKS, BLOCKS+NT)         : per-target flat cell index (int)
//   [BLOCKS+NT, +5*NT)          : per-target contributions {pos,coord,cls,negc,cnt}
constexpr int WS_FLAT  = BLOCKS;
constexpr int WS_CONTR = BLOCKS + NT;

__device__ __forceinline__ float bce0(float x) {        // bce(x, t=0) = softplus(x)
    return fmaxf(x, 0.f) + log1pf(expf(-fabsf(x)));
}
__device__ __forceinline__ float bce1(float x) {        // bce(x, t=1) = softplus(-x)
    return fmaxf(x, 0.f) - x + log1pf(expf(-fabsf(x)));
}

// ---------------------------------------------------------------------------
// Kernel A: streaming reduction of softplus(obj_logit) over all cells.
// Deterministic: one partial per block via LDS tree; no float atomics.
// ---------------------------------------------------------------------------
__global__ __launch_bounds__(TPB)
void yolo_objneg_partials(const float* __restrict__ preds, float* __restrict__ ws) {
    const int g      = blockIdx.x * TPB + threadIdx.x;
    const int stride = BLOCKS * TPB;
    float s = 0.f;
    for (int c = g; c < Mcells; c += stride) {
        const float* p = preds + (size_t)c * Cq;
        __builtin_prefetch(preds + (size_t)(c + stride) * Cq, 0, 0);  // global_prefetch_b8
        s += bce0(p[0]);
    }
    __shared__ float red[TPB];
    red[threadIdx.x] = s;
    __syncthreads();
    for (int off = TPB >> 1; off > 0; off >>= 1) {
        if (threadIdx.x < off) red[threadIdx.x] += red[threadIdx.x + off];
        __syncthreads();
    }
    if (threadIdx.x == 0) ws[blockIdx.x] = red[0];
}

// ---------------------------------------------------------------------------
// Per-target geometry
// ---------------------------------------------------------------------------
struct TgtInfo {
    int   flat;           // cell index or Mcells if invalid
    float gx, gy; int gi, gj;
    float bw, bh, awb, ahb;
};

__device__ __forceinline__ TgtInfo tgt_info(int t, const float* __restrict__ targets,
                                            const float* __restrict__ anc) {
    TgtInfo r;
    const int b = t / Nq;
    const float cx = targets[t * 4 + 0], cy = targets[t * 4 + 1];
    r.bw = targets[t * 4 + 2]; r.bh = targets[t * 4 + 3];
    r.gx = cx * (float)Wq;  r.gy = cy * (float)Hq;
    r.gi = (int)floorf(r.gx); r.gj = (int)floorf(r.gy);
    const bool valid = (r.gi >= 0) && (r.gi < Wq) && (r.gj >= 0) && (r.gj < Hq);
    // first-max argmax over 3 anchors of inter/union (matches jnp.argmax)
    const float area = r.bw * r.bh;
    int ba = 0; float best = -1.f; float awb = anc[0], ahb = anc[1];
    #pragma unroll
    for (int a = 0; a < Aq; ++a) {
        const float aw = anc[2 * a], ah = anc[2 * a + 1];
        const float inter = fminf(aw, r.bw) * fminf(ah, r.bh);
        const float ratio = inter / (aw * ah + area - inter + 1e-9f);
        if (ratio > best) { best = ratio; ba = a; awb = aw; ahb = ah; }
    }
    r.awb = awb; r.ahb = ahb;
    const int flat = ((b * Aq + ba) * Hq + r.gj) * Wq + r.gi;
    r.flat = valid ? flat : Mcells;
    return r;
}

// ---------------------------------------------------------------------------
// Kernel T1: per-target flat index -> workspace
// ---------------------------------------------------------------------------
__global__ __launch_bounds__(64)
void yolo_target_flats(const float* __restrict__ targets,
                       const float* __restrict__ anchors,
                       float* __restrict__ ws) {
    const int t = blockIdx.x * 64 + threadIdx.x;
    if (t >= NT) return;
    float anc[6];
    #pragma unroll
    for (int i = 0; i < 6; ++i) anc[i] = anchors[i];
    ((int*)ws)[WS_FLAT + t] = tgt_info(t, targets, anc).flat;
}

// ---------------------------------------------------------------------------
// Kernel T2: winner scan (last target index wins a cell, matching scatter
// .set semantics) + gathered per-target loss terms. Every one of the 5*NT
// contribution slots is written each call (zeros for losers/invalid).
// ---------------------------------------------------------------------------
__global__ __launch_bounds__(64)
void yolo_target_terms(const float* __restrict__ preds,
                       const float* __restrict__ targets,
                       const int*   __restrict__ tcls,
                       const float* __restrict__ anchors,
                       float* __restrict__ ws) {
    const int tid = threadIdx.x;
    const int t   = blockIdx.x * 64 + tid;

    __shared__ int sflat[NT];
    const int* wflat = (const int*)ws + WS_FLAT;
    for (int i = tid; i < NT; i += 64) sflat[i] = wflat[i];
    __syncthreads();
    if (t >= NT) return;

    float pos = 0.f, coord = 0.f, cls = 0.f, negc = 0.f, cnt = 0.f;
    const int flat = sflat[t];
    if (flat < Mcells) {
        bool win = true;
        for (int t2 = t + 1; t2 < NT; ++t2)
            if (sflat[t2] == flat) { win = false; break; }
        if (win) {
            cnt = 1.f;
            float anc[6];
            #pragma unroll
            for (int i = 0; i < 6; ++i) anc[i] = anchors[i];
            const TgtInfo ti = tgt_info(t, targets, anc);
            const float* p = preds + (size_t)flat * Cq;

            const float obj = p[0];
            pos  = bce1(obj);
            negc = bce0(obj);                 // remove this cell from the "neg" sum

            const float sx = 1.f / (1.f + expf(-p[1]));
            const float sy = 1.f / (1.f + expf(-p[2]));
            const float txt = ti.gx - (float)ti.gi;
            const float tyt = ti.gy - (float)ti.gj;
            const float twt = logf(ti.bw / (ti.awb + 1e-9f) + 1e-9f);
            const float tht = logf(ti.bh / (ti.ahb + 1e-9f) + 1e-9f);
            const float dx = sx - txt, dy = sy - tyt, dw = p[3] - twt, dh = p[4] - tht;
            coord = dx * dx + dy * dy + dw * dw + dh * dh;

            int ci = tcls[t];
            ci = ci < 0 ? 0 : (ci > NCq - 1 ? NCq - 1 : ci);
            float mx = -3.0e38f;
            #pragma unroll
            for (int j = 0; j < NCq; ++j) mx = fmaxf(mx, p[5 + j]);
            float se = 0.f;
            #pragma unroll
            for (int j = 0; j < NCq; ++j) se += expf(p[5 + j] - mx);
            const float lse = mx + logf(se);
            cls = -(p[5 + ci] - lse);
        }
    }
    float* contr = ws + WS_CONTR;
    contr[0 * NT + t] = pos;
    contr[1 * NT + t] = coord;
    contr[2 * NT + t] = cls;
    contr[3 * NT + t] = negc;
    contr[4 * NT + t] = cnt;
}

// ---------------------------------------------------------------------------
// Exact fp32 sum of n64*64 floats on one wave via v_wmma_f32_16x16x4_f32:
// A = ones(16x4) -> D[m][n] = sum_k B[k][n] + C[m][n]; accumulating through C
// sums every B entry into the (row-identical) column sums. Summing element 0
// of the v8f over all 32 lanes counts rows 0 and 8 (identical) -> 2x total.
// ---------------------------------------------------------------------------
__device__ __forceinline__ float wmma_sum(const float* __restrict__ p, int n64, int lane) {
    v2f ones; ones.x = 1.f; ones.y = 1.f;
    v8f acc = {};
    for (int i = 0; i < n64; ++i) {
        v2f bm;
        bm.x = p[i * 64 + lane];
        bm.y = p[i * 64 + 32 + lane];
        acc = __builtin_amdgcn_wmma_f32_16x16x4_f32(
            false, ones, false, bm, (short)0, acc, false, false);
    }
    float s = acc[0];
    #pragma unroll
    for (int m = 1; m < 32; m <<= 1) s += __shfl_xor(s, m, 32);
    return 0.5f * s;
}

// ---------------------------------------------------------------------------
// Kernel F: single wave32 (EXEC all-ones). WMMA-reduce the block partials and
// the five contribution arrays, then emit the scalar loss.
// ---------------------------------------------------------------------------
__global__ __launch_bounds__(32)
void yolo_finalize(const float* __restrict__ ws, float* __restrict__ out) {
    const int lane = threadIdx.x;
    const float s_all = wmma_sum(ws,                     CHUNKS_P, lane);
    const float pos   = wmma_sum(ws + WS_CONTR + 0 * NT, CHUNKS_T, lane);
    const float coord = wmma_sum(ws + WS_CONTR + 1 * NT, CHUNKS_T, lane);
    const float cls   = wmma_sum(ws + WS_CONTR + 2 * NT, CHUNKS_T, lane);
    const float negc  = wmma_sum(ws + WS_CONTR + 3 * NT, CHUNKS_T, lane);
    const float cnt   = wmma_sum(ws + WS_CONTR + 4 * NT, CHUNKS_T, lane);

    if (lane == 0) {
        const float has   = (cnt > 0.f) ? 1.f : 0.f;
        const float total = has * (L_COORD * coord + pos + L_CLS * cls)
                          + L_NOOBJ * (s_all - negc);
        out[0] = total / (float)Bq;
    }
}

// ---------------------------------------------------------------------------
extern "C" void kernel_launch(void* const* d_in, const int* in_sizes, int n_in,
                              void* d_out, int out_size, void* d_ws, size_t ws_size,
                              hipStream_t stream) {
    const float* preds   = (const float*)d_in[0];
    const float* targets = (const float*)d_in[1];
    const int*   tcls    = (const int*)  d_in[2];
    const float* anchors = (const float*)d_in[3];
    float* ws  = (float*)d_ws;               // (BLOCKS + NT + 5*NT) floats of scratch
    float* out = (float*)d_out;

    yolo_objneg_partials<<<BLOCKS, TPB, 0, stream>>>(preds, ws);
    yolo_target_flats  <<<(NT + 63) / 64, 64, 0, stream>>>(targets, anchors, ws);
    yolo_target_terms  <<<(NT + 63) / 64, 64, 0, stream>>>(preds, targets, tcls, anchors, ws);
    yolo_finalize      <<<1, 32, 0, stream>>>(ws, out);
}